// Ricker_84988812853296
// MI455X (gfx1250) — compile-verified
//
#include <hip/hip_runtime.h>

// Ricker recurrence scan on gfx1250: one row (chain) per lane.
// Serial chain per step: v_fma -> v_exp_f32 -> v_mul (log2-domain folded).
// Memory: 8-deep circular GLOBAL_LOAD_ASYNC_TO_LDS_B128 staging per wave
// (ASYNCcnt-tracked, wave-private buffers, no barriers, ~7-chunk prefetch
// distance to cover HBM latency). Fallback: direct B128 + global_prefetch_b8.

#define T_LEN 4096
#define CH    16               // steps per chunk: CH*8B = 128B per lane per chunk
#define VECS  (CH / 2)         // 8 float4 per chunk per lane
#define BLOCK 64
#define WAVES (BLOCK / 32)
#define NBUF  8                // circular staging depth (power of two)

#if __has_builtin(__builtin_amdgcn_global_load_async_to_lds_b128)
#define HAVE_ASYNC_LDS 1
#else
#define HAVE_ASYNC_LDS 0
#endif

typedef int v4i __attribute__((ext_vector_type(4)));
typedef __attribute__((address_space(1))) v4i* as1_v4ip;   // global src
typedef __attribute__((address_space(3))) v4i* as3_v4ip;   // LDS dst

__device__ __forceinline__ float fast_exp2(float x) {
#if __has_builtin(__builtin_amdgcn_exp2f)
    return __builtin_amdgcn_exp2f(x);   // v_exp_f32
#else
    return exp2f(x);
#endif
}

__device__ __forceinline__ float qnan_f32() {
    union { unsigned u; float f; } v; v.u = 0x7fc00000u; return v.f;
}

template<int N>
__device__ __forceinline__ void wait_asynccnt() {
#if __has_builtin(__builtin_amdgcn_s_wait_asynccnt)
    __builtin_amdgcn_s_wait_asynccnt(N);
#else
    asm volatile("s_wait_asynccnt %0" :: "i"(N) : "memory");
#endif
}

// Tail wait: threshold must be a compile-time immediate -> constant switch.
__device__ __forceinline__ void wait_async_thr(int thr) {
    switch (thr) {
        case 0:          wait_asynccnt<0>();          break;
        case 1 * VECS:   wait_asynccnt<1 * VECS>();   break;
        case 2 * VECS:   wait_asynccnt<2 * VECS>();   break;
        case 3 * VECS:   wait_asynccnt<3 * VECS>();   break;
        case 4 * VECS:   wait_asynccnt<4 * VECS>();   break;
        case 5 * VECS:   wait_asynccnt<5 * VECS>();   break;
        default:         wait_asynccnt<6 * VECS>();   break;
    }
}

// ---- shared compute: unpack chunk, covariate precompute, serial exp chain ----
template<bool REC>
__device__ __forceinline__ void compute_store_chunk(int k, const float4* q, float& r,
                                                    float al, float bl, float dl, float gl,
                                                    float p4, float4* __restrict__ orow) {
    float s[CH], u[CH], c[CH], ret[CH];
    #pragma unroll
    for (int m = 0; m < VECS; ++m) {
        s[2 * m]     = q[m].x;  u[2 * m]     = q[m].y;
        s[2 * m + 1] = q[m].z;  u[2 * m + 1] = q[m].w;
    }
    // Off-chain: c_j = (alpha + d*u + g*u^2) * log2e
    #pragma unroll
    for (int j = 0; j < CH; ++j)
        c[j] = __builtin_fmaf(u[j], __builtin_fmaf(gl, u[j], dl), al);

    if (k == 0) {
        if (REC) { r = s[0]; }
        else     { r = s[0] * fast_exp2(__builtin_fmaf(-bl, s[0], c[0])); }
        ret[0] = qnan_f32();
    }
    #pragma unroll
    for (int j = 0; j < CH; ++j) {
        if (j == 0 && k == 0) continue;          // step 0 handled above
        float base = REC ? r : s[j];
        float e  = fast_exp2(__builtin_fmaf(-bl, base, c[j]));
        float rn = base * e;
        ret[j] = (rn + r) * p4;
        r = rn;
    }
    #pragma unroll
    for (int m = 0; m < CH / 4; ++m)
        orow[(CH / 4) * k + m] =
            make_float4(ret[4 * m], ret[4 * m + 1], ret[4 * m + 2], ret[4 * m + 3]);
}

#if HAVE_ASYNC_LDS
// Issue one chunk's async global->LDS batch (8 x B128 per lane).
// Buffer layout: [vec j][lane] float4 -> contiguous 512B sweep per j.
__device__ __forceinline__ void issue_stage(const float4* __restrict__ gsrc,
                                            float4* lds_lane_base) {
    #pragma unroll
    for (int j = 0; j < VECS; ++j) {
        __builtin_amdgcn_global_load_async_to_lds_b128(
            (as1_v4ip)(gsrc + j),
            (as3_v4ip)(lds_lane_base + j * 32),
            /*offset=*/0, /*cpol=*/0);
    }
}

template<bool REC>
__device__ __forceinline__ void ricker_row(const float* __restrict__ x,
                                           float* __restrict__ out,
                                           int row, int T,
                                           float al, float bl, float dl, float gl, float p4,
                                           float4* wbuf /* lane base of buffer 0 */) {
    const float4* __restrict__ xr   = (const float4*)(x + (size_t)row * (size_t)T * 2u);
    float4* __restrict__       orow = (float4*)(out + (size_t)row * (size_t)T);
    const int nch = T / CH;
    float r = 0.0f;

    // Prologue: fill NBUF-1 buffers (56 async loads outstanding, counter max 63)
    const int npro = (nch < NBUF - 1) ? nch : (NBUF - 1);
    for (int b = 0; b < npro; ++b)
        issue_stage(xr + VECS * b, wbuf + (size_t)b * (VECS * 32));

    int k = 0;
    // Steady state: wait (batch k landed; loads complete in order), refill, compute
    for (; k + (NBUF - 1) < nch; ++k) {
        wait_asynccnt<(NBUF - 2) * VECS>();
        issue_stage(xr + VECS * (k + NBUF - 1),
                    wbuf + (size_t)((k + NBUF - 1) & (NBUF - 1)) * (VECS * 32));
        asm volatile("" ::: "memory");
        float4 q[VECS];
        float4* cur = wbuf + (size_t)(k & (NBUF - 1)) * (VECS * 32);
        #pragma unroll
        for (int j = 0; j < VECS; ++j) q[j] = cur[j * 32];   // ds_load_b128
        compute_store_chunk<REC>(k, q, r, al, bl, dl, gl, p4, orow);
    }
    // Tail: drain remaining buffers
    for (; k < nch; ++k) {
        int rem = nch - 1 - k;
        if (rem > NBUF - 2) rem = NBUF - 2;
        wait_async_thr(rem * VECS);
        asm volatile("" ::: "memory");
        float4 q[VECS];
        float4* cur = wbuf + (size_t)(k & (NBUF - 1)) * (VECS * 32);
        #pragma unroll
        for (int j = 0; j < VECS; ++j) q[j] = cur[j * 32];
        compute_store_chunk<REC>(k, q, r, al, bl, dl, gl, p4, orow);
    }
}
#else
template<bool REC>
__device__ __forceinline__ void ricker_row(const float* __restrict__ x,
                                           float* __restrict__ out,
                                           int row, int T,
                                           float al, float bl, float dl, float gl, float p4) {
    const float4* __restrict__ xr   = (const float4*)(x + (size_t)row * (size_t)T * 2u);
    float4* __restrict__       orow = (float4*)(out + (size_t)row * (size_t)T);
    const int nch = T / CH;
    float r = 0.0f;
    for (int k = 0; k < nch; ++k) {
        if (k + 1 < nch)
            __builtin_prefetch((const void*)(xr + VECS * (k + 1)), 0, 0);
        float4 q[VECS];
        #pragma unroll
        for (int m = 0; m < VECS; ++m) q[m] = xr[VECS * k + m];
        compute_store_chunk<REC>(k, q, r, al, bl, dl, gl, p4, orow);
    }
}
#endif

__global__ __launch_bounds__(BLOCK) void ricker_scan_kernel(
        const float* __restrict__ x,
        const float* __restrict__ params,
        const float* __restrict__ p4p,
        const int*   __restrict__ recp,
        float* __restrict__ out,
        int B, int T) {
#if HAVE_ASYNC_LDS
    __shared__ float4 stage[WAVES * NBUF * VECS * 32];   // 64 KB per block
#endif
    int row = blockIdx.x * blockDim.x + threadIdx.x;
    if (row >= B) return;

    const float L2E = 1.4426950408889634f;
    float alpha = params[0], beta = params[1], d = params[2], g = params[3];
    float p4 = p4p[0];
    float al = alpha * L2E, bl = beta * L2E, dl = d * L2E, gl = g * L2E;

#if HAVE_ASYNC_LDS
    int wave = threadIdx.x >> 5, lane = threadIdx.x & 31;
    float4* wbuf = &stage[(size_t)wave * (NBUF * VECS * 32) + lane];
    if (recp[0] != 0)
        ricker_row<true >(x, out, row, T, al, bl, dl, gl, p4, wbuf);
    else
        ricker_row<false>(x, out, row, T, al, bl, dl, gl, p4, wbuf);
#else
    if (recp[0] != 0)
        ricker_row<true >(x, out, row, T, al, bl, dl, gl, p4);
    else
        ricker_row<false>(x, out, row, T, al, bl, dl, gl, p4);
#endif
}

extern "C" void kernel_launch(void* const* d_in, const int* in_sizes, int n_in,
                              void* d_out, int out_size, void* d_ws, size_t ws_size,
                              hipStream_t stream) {
    (void)n_in; (void)d_ws; (void)ws_size; (void)out_size;
    const float* x      = (const float*)d_in[0];
    const float* params = (const float*)d_in[1];
    const float* p4     = (const float*)d_in[2];
    const int*   rec    = (const int*)d_in[3];
    float*       out    = (float*)d_out;

    const int T = T_LEN;
    const int B = in_sizes[0] / (2 * T);   // x is [B, T, 2]

    dim3 block(BLOCK);
    dim3 grid((B + BLOCK - 1) / BLOCK);
    ricker_scan_kernel<<<grid, block, 0, stream>>>(x, params, p4, rec, out, B, T);
}